// FD_discretizer_90134183674492
// MI455X (gfx1250) — compile-verified
//
#include <hip/hip_runtime.h>
#include <cstdint>

// ---------------------------------------------------------------------------
// FD discretizer for 1024x1024 structured grid (extended 1026x1026).
// Kernel A: build BC-enforced extended fields (padded SoA planes in d_ws).
// Kernel B: fused stencil pass producing all four outputs, staging tiles
//           into LDS with CDNA5 GLOBAL_LOAD_ASYNC_TO_LDS_B128/B32 (ASYNCcnt).
// ---------------------------------------------------------------------------

#define NXc 1024
#define NYc 1024
#define Exc (NXc + 2)          // 1026
#define Eyc (NYc + 2)          // 1026
#define Nn  (NXc * NYc)
#define WSP 1028               // padded plane row stride (4112 B, 16B aligned)
#define NPLANE (Eyc * WSP)     // floats per workspace plane
#define TLS 36                 // LDS tile row stride (144 B, 16B aligned)
#define TYs 10                 // tile height (8 centers + 2 halo)

enum NodeType { T_NORMAL = 0, T_INFLOW = 4, T_OUTFLOW = 5, T_WALL = 6, T_PRESS = 7 };

typedef __attribute__((__vector_size__(4 * sizeof(int)))) int v4i;
typedef __attribute__((address_space(1))) int  gint_t;
typedef __attribute__((address_space(3))) int  lint_t;
typedef __attribute__((address_space(1))) v4i  gv4i_t;
typedef __attribute__((address_space(3))) v4i  lv4i_t;

// ---------------- CDNA5 async global -> LDS copies --------------------------
__device__ __forceinline__ void async_ld_b32(void* ldsDst, const float* base, int elem) {
#if __has_builtin(__builtin_amdgcn_global_load_async_to_lds_b32)
  __builtin_amdgcn_global_load_async_to_lds_b32(
      (gint_t*)(uintptr_t)(base + elem),
      (lint_t*)(uint32_t)(uintptr_t)ldsDst, 0, 0);
#else
  unsigned l = (unsigned)(uintptr_t)ldsDst;
  int off = elem * 4;
  asm volatile("global_load_async_to_lds_b32 %0, %1, %2"
               :: "v"(l), "v"(off), "s"(base) : "memory");
#endif
}

__device__ __forceinline__ void async_ld_b128(void* ldsDst, const float* base, int elem) {
#if __has_builtin(__builtin_amdgcn_global_load_async_to_lds_b128)
  __builtin_amdgcn_global_load_async_to_lds_b128(
      (gv4i_t*)(uintptr_t)(base + elem),
      (lv4i_t*)(uint32_t)(uintptr_t)ldsDst, 0, 0);
#else
  async_ld_b32(ldsDst, base, elem);
  async_ld_b32((char*)ldsDst + 4,  base, elem + 1);
  async_ld_b32((char*)ldsDst + 8,  base, elem + 2);
  async_ld_b32((char*)ldsDst + 12, base, elem + 3);
#endif
}

__device__ __forceinline__ void wait_async0() {
#if __has_builtin(__builtin_amdgcn_s_wait_asynccnt)
  __builtin_amdgcn_s_wait_asynccnt(0);
#else
  asm volatile("s_wait_asynccnt 0" ::: "memory");
#endif
}

// ---------------------------------------------------------------------------
// Kernel A: extended fields + hard BC (ghost extrapolation) + press point.
// 2D launch: x = extended column (ie), y = extended row (je). No div/mod.
// ---------------------------------------------------------------------------
__global__ __launch_bounds__(256) void build_ext_kernel(
    const float* __restrict__ orig, const float* __restrict__ uold,
    const float* __restrict__ nodey, const int* __restrict__ ntype,
    float* __restrict__ eu, float* __restrict__ ev, float* __restrict__ ep,
    float* __restrict__ ou, float* __restrict__ ov)
{
  const int ie = blockIdx.x * 256 + threadIdx.x;
  const int je = blockIdx.y;
  if (ie >= Exc) return;

  int io = ie - 1; io = io < 0 ? 0 : (io > NXc - 1 ? NXc - 1 : io);
  int jo = je - 1; jo = jo < 0 ? 0 : (jo > NYc - 1 ? NYc - 1 : jo);
  const int m = jo * NXc + io;

  float u = orig[3 * m], v = orig[3 * m + 1], p = orig[3 * m + 2];
  float uo = uold[3 * m], vo = uold[3 * m + 1];

  // boundary ring (corners excluded) — matches ghost_stencil construction
  int side = 0;
  if      (ie == 0       && je >= 1 && je <= Eyc - 2) side = 1; // INFLOW
  else if (ie == Exc - 1 && je >= 1 && je <= Eyc - 2) side = 2; // OUTFLOW
  else if (je == 0       && ie >= 1 && ie <= Exc - 2) side = 3; // WALL
  else if (je == Eyc - 1 && ie >= 1 && ie <= Exc - 2) side = 4; // WALL

  if (side) {
    int i1, j1, i2, j2;
    if      (side == 1) { i1 = 1;       j1 = je;      i2 = 2;       j2 = je;      }
    else if (side == 2) { i1 = Exc - 2; j1 = je;      i2 = Exc - 3; j2 = je;      }
    else if (side == 3) { i1 = ie;      j1 = 1;       i2 = ie;      j2 = 2;       }
    else                { i1 = ie;      j1 = Eyc - 2; i2 = ie;      j2 = Eyc - 3; }
    const int m1 = (j1 - 1) * NXc + (i1 - 1);  // g1,g2 are interior ext nodes
    const int m2 = (j2 - 1) * NXc + (i2 - 1);
    const int nt1 = ntype[m1];
    const bool bc1 = (nt1 == T_INFLOW) || (nt1 == T_WALL);

    // "dummy" values at g1 (per reference _hard_enforce_bc), new field
    const float du = bc1 ? nodey[3 * m1]     : orig[3 * m1];
    const float dv = bc1 ? nodey[3 * m1 + 1] : orig[3 * m1 + 1];
    const float dp = (nt1 == T_OUTFLOW) ? 0.0f : orig[3 * m1 + 2];
    const float g2u = orig[3 * m2], g2v = orig[3 * m2 + 1], g2p = orig[3 * m2 + 2];
    if (side == 2) { u = g2u; v = g2v; p = 2.0f * dp - g2p; }             // uvmask
    else           { u = 2.0f * du - g2u; v = 2.0f * dv - g2v; p = g2p; } // pmask

    // old field (only uv needed downstream)
    const float duo = bc1 ? nodey[3 * m1]     : uold[3 * m1];
    const float dvo = bc1 ? nodey[3 * m1 + 1] : uold[3 * m1 + 1];
    const float g2uo = uold[3 * m2], g2vo = uold[3 * m2 + 1];
    if (side == 2) { uo = g2uo; vo = g2vo; }
    else           { uo = 2.0f * duo - g2uo; vo = 2.0f * dvo - g2vo; }
  }
  if (je == Eyc / 2 && ie == Exc / 2) p = 0.0f;  // press point (new field only)

  const int e = je * WSP + ie;
  eu[e] = u; ev[e] = v; ep[e] = p; ou[e] = uo; ov[e] = vo;
}

// ---------------------------------------------------------------------------
// Kernel B: fused stencil pass. block = (32,8); each block stages a 34x10
// halo tile of 10 planes into LDS via async-to-LDS (b128 for padded field
// planes, b32 for AoS metrics), then computes:
//   loss_cont, momentum u/v, uvp_to_vis (3x3 weighted average).
// ---------------------------------------------------------------------------
__global__ __launch_bounds__(256) void fd_main_kernel(
    const float* __restrict__ eu, const float* __restrict__ ev, const float* __restrict__ ep,
    const float* __restrict__ ou, const float* __restrict__ ov,
    const float* __restrict__ ebm,
    const float* __restrict__ orig, const float* __restrict__ uold,
    const float* __restrict__ obm,
    const float* __restrict__ dtg, const float* __restrict__ theta,
    const float* __restrict__ relaxp,
    float* __restrict__ out)
{
  __shared__ float sm[10][TYs * TLS];  // eu ev ep ou ov dxx dxy dex dey Jm

  const int tid = threadIdx.y * 32 + threadIdx.x;
  const int ib0 = blockIdx.x * 32;     // halo start column in extended grid
  const int jb0 = blockIdx.y * 8;      // halo start row    in extended grid

  const float* pb[5] = { eu, ev, ep, ou, ov };

  // field planes: 5 planes x 10 rows x 8 b128 chunks (32 of 34 columns)
  for (int t = tid; t < 400; t += 256) {
    const int pl = t / 80, rem = t % 80;
    const int ty = rem / 8, ch = rem % 8;
    const int ge = (jb0 + ty) * WSP + ib0 + 4 * ch;
    async_ld_b128(&sm[pl][ty * TLS + 4 * ch], pb[pl], ge);
  }
  // field plane tails: columns 32,33 of each row
  for (int t = tid; t < 100; t += 256) {
    const int pl = t / 20, rem = t % 20;
    const int ty = rem / 2, k = rem % 2;
    const int ge = (jb0 + ty) * WSP + ib0 + 32 + k;
    async_ld_b32(&sm[pl][ty * TLS + 32 + k], pb[pl], ge);
  }
  // metrics (AoS, 5 floats per ext node) -> SoA planes 5..9
  for (int t = tid; t < 1700; t += 256) {
    const int node = t / 5, k = t % 5;
    const int tx = node % 34, ty = node / 34;
    const int g = (jb0 + ty) * Exc + (ib0 + tx);
    async_ld_b32(&sm[5 + k][ty * TLS + tx], ebm, 5 * g + k);
  }
  wait_async0();
  __syncthreads();

  const int io = ib0 + threadIdx.x;
  const int jo = jb0 + threadIdx.y;
  const int o  = jo * NXc + io;
  const int lc = (threadIdx.y + 1) * TLS + (threadIdx.x + 1);

  const float dt    = dtg[0];
  const float uc    = theta[0], cc = theta[1], convc = theta[2];
  const float pcth  = theta[3], dcth = theta[4];
  const float relax = relaxp[0];

  // 5-point stencil: C, L, R, D, U
  const int idxs[5] = { lc, lc - 1, lc + 1, lc - TLS, lc + TLS };
  float U[5], V[5], Uo[5], Vo[5], a11[5], a22[5];
  float pxx[5], pxy[5], pex[5], pey[5];
  float uu[5], vv[5], uuo[5], vvo[5];
#pragma unroll
  for (int s = 0; s < 5; ++s) {
    const int t = idxs[s];
    const float u_  = sm[0][t], v_  = sm[1][t], p_ = sm[2][t];
    const float uo_ = sm[3][t], vo_ = sm[4][t];
    const float mxx = sm[5][t], mxy = sm[6][t];
    const float mex = sm[7][t], mey = sm[8][t];
    const float invJ = 1.0f / sm[9][t];
    uu[s] = u_; vv[s] = v_; uuo[s] = uo_; vvo[s] = vo_;
    U[s]  = (u_ * mxx + v_ * mxy) * invJ;
    V[s]  = (u_ * mex + v_ * mey) * invJ;
    Uo[s] = (uo_ * mxx + vo_ * mxy) * invJ;
    Vo[s] = (uo_ * mex + vo_ * mey) * invJ;
    a11[s] = (mxx * mxx + mxy * mxy) * invJ;
    a22[s] = (mex * mex + mey * mey) * invJ;
    const float pq = p_ * invJ;
    pxx[s] = pq * mxx; pxy[s] = pq * mxy;
    pex[s] = pq * mex; pey[s] = pq * mey;
  }
  enum { C = 0, L = 1, R = 2, D = 3, Up = 4 };

  // continuity
  const float UfL = 0.5f * (U[L] + U[C]), UfR = 0.5f * (U[C] + U[R]);
  const float VfD = 0.5f * (V[D] + V[C]), VfU = 0.5f * (V[C] + V[Up]);
  const float loss = (UfR - UfL + VfU - VfD) * cc;

  // convection (new)
  const float uxiL = 0.5f * (uu[L] + uu[C]), uxiR = 0.5f * (uu[C] + uu[R]);
  const float vxiL = 0.5f * (vv[L] + vv[C]), vxiR = 0.5f * (vv[C] + vv[R]);
  const float uetD = 0.5f * (uu[D] + uu[C]), uetU = 0.5f * (uu[C] + uu[Up]);
  const float vetD = 0.5f * (vv[D] + vv[C]), vetU = 0.5f * (vv[C] + vv[Up]);
  const float d1u_n = uxiR * UfR - uxiL * UfL, d1v_n = vxiR * UfR - vxiL * UfL;
  const float d2u_n = uetU * VfU - uetD * VfD, d2v_n = vetU * VfU - vetD * VfD;

  // convection (old)
  const float UofL = 0.5f * (Uo[L] + Uo[C]), UofR = 0.5f * (Uo[C] + Uo[R]);
  const float VofD = 0.5f * (Vo[D] + Vo[C]), VofU = 0.5f * (Vo[C] + Vo[Up]);
  const float uxiLo = 0.5f * (uuo[L] + uuo[C]), uxiRo = 0.5f * (uuo[C] + uuo[R]);
  const float vxiLo = 0.5f * (vvo[L] + vvo[C]), vxiRo = 0.5f * (vvo[C] + vvo[R]);
  const float uetDo = 0.5f * (uuo[D] + uuo[C]), uetUo = 0.5f * (uuo[C] + uuo[Up]);
  const float vetDo = 0.5f * (vvo[D] + vvo[C]), vetUo = 0.5f * (vvo[C] + vvo[Up]);
  const float d1u_o = uxiRo * UofR - uxiLo * UofL, d1v_o = vxiRo * UofR - vxiLo * UofL;
  const float d2u_o = uetUo * VofU - uetDo * VofD, d2v_o = vetUo * VofU - vetDo * VofD;

  const float w = 1.0f - relax;
  const float d1u = d1u_o * relax + d1u_n * w;
  const float d1v = d1v_o * relax + d1v_n * w;
  const float d2u = d2u_o * relax + d2u_n * w;
  const float d2v = d2v_o * relax + d2v_n * w;

  // diffusion (new)
  const float a11L = 0.5f * (a11[L] + a11[C]), a11R = 0.5f * (a11[C] + a11[R]);
  const float a22D = 0.5f * (a22[D] + a22[C]), a22U = 0.5f * (a22[C] + a22[Up]);
  const float dv1u = a11R * (uu[R] - uu[C]) - a11L * (uu[C] - uu[L]);
  const float dv1v = a11R * (vv[R] - vv[C]) - a11L * (vv[C] - vv[L]);
  const float dv2u = a22U * (uu[Up] - uu[C]) - a22D * (uu[C] - uu[D]);
  const float dv2v = a22U * (vv[Up] - vv[C]) - a22D * (vv[C] - vv[D]);

  // pressure gradient
  const float gPx = 0.5f * (pxx[C] + pxx[R]) - 0.5f * (pxx[L] + pxx[C])
                  + 0.5f * (pex[C] + pex[Up]) - 0.5f * (pex[D] + pex[C]);
  const float gPy = 0.5f * (pey[C] + pey[Up]) - 0.5f * (pey[D] + pey[C])
                  + 0.5f * (pxy[C] + pxy[R]) - 0.5f * (pxy[L] + pxy[C]);

  // unsteady
  const float o_u = orig[3 * o], o_v = orig[3 * o + 1];
  const float l_u = uold[3 * o], l_v = uold[3 * o + 1];
  const float Jo  = obm[5 * o + 4];
  const float unsu = (o_u - l_u) / dt / Jo;
  const float unsv = (o_v - l_v) / dt / Jo;

  const float momu = uc * unsu + convc * (d1u + d2u) + pcth * gPx - dcth * (dv1u + dv2u);
  const float momv = uc * unsv + convc * (d1v + d2v) + pcth * gPy - dcth * (dv1v + dv2v);

  // uvp_to_vis: average of 4 overlapping node-cells (3x3 weighted stencil)
  float vis[3];
#pragma unroll
  for (int cpl = 0; cpl < 3; ++cpl) {
    const float* P = sm[cpl];
    const int b = lc;
    const float eLL = P[b - TLS - 1], eCL = P[b - TLS], eRL = P[b - TLS + 1];
    const float eLC = P[b - 1],       eCC = P[b],       eRC = P[b + 1];
    const float eLU = P[b + TLS - 1], eCU = P[b + TLS], eRU = P[b + TLS + 1];
    const float cellDL = 0.25f * (eLL + eCL + eLC + eCC); // cell (ie0-1, je0-1)
    const float cellDR = 0.25f * (eCL + eRL + eCC + eRC); // cell (ie0,   je0-1)
    const float cellUL = 0.25f * (eLC + eCC + eLU + eCU); // cell (ie0-1, je0)
    const float cellUR = 0.25f * (eCC + eRC + eCU + eRU); // cell (ie0,   je0)
    const float faceU = 0.5f * (cellUL + cellUR);
    const float faceD = 0.5f * (cellDL + cellDR);
    vis[cpl] = 0.5f * (faceU + faceD);
  }

  out[o]          = loss;
  out[Nn + o]     = momu;
  out[2 * Nn + o] = momv;
  float* vo_ = out + 3 * Nn + 3 * o;
  vo_[0] = vis[0]; vo_[1] = vis[1]; vo_[2] = vis[2];
}

// ---------------------------------------------------------------------------
extern "C" void kernel_launch(void* const* d_in, const int* in_sizes, int n_in,
                              void* d_out, int out_size, void* d_ws, size_t ws_size,
                              hipStream_t stream) {
  (void)in_sizes; (void)n_in; (void)out_size; (void)ws_size;
  const float* orig  = (const float*)d_in[0];   // original_uv  (n,3)
  const float* uold  = (const float*)d_in[1];   // uv_old       (n,3)
  const float* nodey = (const float*)d_in[2];   // node_y       (n,3)
  const float* obm   = (const float*)d_in[3];   // original_block_metrics (n,5)
  const float* ebm   = (const float*)d_in[4];   // extended_block_metrics (n_ext,5)
  const float* dtg   = (const float*)d_in[5];   // dt_graph (1,1)
  const float* theta = (const float*)d_in[6];   // pde_theta (1,5)
  const float* relax = (const float*)d_in[7];   // relaxtion (1,1)
  const int*   ntype = (const int*)d_in[8];     // node_type (n,)
  // index tables d_in[9..18] unused: regular grid -> analytic indexing

  float* ws = (float*)d_ws;                     // needs 5*NPLANE*4 ~= 21.1 MB
  float* eu = ws;
  float* ev = ws + (size_t)NPLANE;
  float* ep = ws + (size_t)2 * NPLANE;
  float* ou = ws + (size_t)3 * NPLANE;
  float* ov = ws + (size_t)4 * NPLANE;

  dim3 blkA(256, 1, 1);
  dim3 grdA((Exc + 255) / 256, Eyc, 1);
  build_ext_kernel<<<grdA, blkA, 0, stream>>>(
      orig, uold, nodey, ntype, eu, ev, ep, ou, ov);

  dim3 blkB(32, 8, 1);
  dim3 grdB(NXc / 32, NYc / 8, 1);
  fd_main_kernel<<<grdB, blkB, 0, stream>>>(
      eu, ev, ep, ou, ov, ebm, orig, uold, obm, dtg, theta, relax, (float*)d_out);
}